// Model_67774583931486
// MI455X (gfx1250) — compile-verified
//
#include <hip/hip_runtime.h>

typedef __attribute__((ext_vector_type(2))) float v2f;
typedef __attribute__((ext_vector_type(8))) float v8f;

#define N_NODES 100000
#define D 256
#define O 128
#define E_MSG 1600000
#define E_TGT 10000
#define RRELU_SLOPE 0.2291666666666667f

// ---------------- zero the agg accumulator (workspace) ----------------
__global__ void zero_kernel(float4* __restrict__ p, int n4) {
    int i = blockIdx.x * blockDim.x + threadIdx.x;
    if (i < n4) p[i] = make_float4(0.f, 0.f, 0.f, 0.f);
}

// ---------------- edge scatter: agg[dst] += w * x[src] ----------------
// 4 edges per 256-thread block; 64 lanes per edge, float4 per lane.
__global__ void scatter_kernel(const float* __restrict__ x,
                               const int* __restrict__ mei,
                               const int* __restrict__ tei,
                               const float* __restrict__ tw,
                               float* __restrict__ agg) {
    const int tid  = threadIdx.x;
    const int sub  = tid >> 6;        // edge within block
    const int lane = tid & 63;        // 64 lanes cover 256 floats as float4
    const int e    = blockIdx.x * 4 + sub;
    const int E    = E_MSG + E_TGT;
    if (e >= E) return;

    int src, dst; float w;
    if (e < E_MSG) {
        src = mei[e];
        dst = mei[E_MSG + e];
        w   = 1.0f;
    } else {
        int j = e - E_MSG;
        // target edges are flipped: src = tei[1][j], dst = tei[0][j]
        src = tei[E_TGT + j];
        dst = tei[j];
        w   = tw[j];
    }

    const float4 xv = *reinterpret_cast<const float4*>(x + (size_t)src * D + lane * 4);
    float* a = agg + (size_t)dst * D + lane * 4;
    __hip_atomic_fetch_add(a + 0, xv.x * w, __ATOMIC_RELAXED, __HIP_MEMORY_SCOPE_AGENT);
    __hip_atomic_fetch_add(a + 1, xv.y * w, __ATOMIC_RELAXED, __HIP_MEMORY_SCOPE_AGENT);
    __hip_atomic_fetch_add(a + 2, xv.z * w, __ATOMIC_RELAXED, __HIP_MEMORY_SCOPE_AGENT);
    __hip_atomic_fetch_add(a + 3, xv.w * w, __ATOMIC_RELAXED, __HIP_MEMORY_SCOPE_AGENT);
}

// ---------------- fused GEMM1 + relu + GEMM2 + relu/rrelu ----------------
// block: 256 threads = 8 wave32; 32 output rows per block.
// Wave w owns cols [16w,16w+16) and computes TWO 16x16 row tiles, sharing each
// B fragment between both WMMAs (halves W traffic vs 16-row blocks).
__global__ __launch_bounds__(256) void fused_gemm_kernel(
    const float* __restrict__ agg, const float* __restrict__ x,
    const float* __restrict__ Wrel, const float* __restrict__ brel,
    const float* __restrict__ Wroot, const float* __restrict__ Wmu,
    const float* __restrict__ bmu, float* __restrict__ out) {
    constexpr int LDA = 516;   // dword stride: mult of 4 (b128), ≡4 mod 64 (bank-conflict-free b64 reads)
    constexpr int LDH = 132;   // same properties for the h tile
    __shared__ float At[32 * LDA];   // 32 x 512 tile of [agg | x]  (~66 KB)
    __shared__ float Ht[32 * LDH];   // 32 x 128 tile of h          (~17 KB)

    const int tid     = threadIdx.x;
    const int rowBase = blockIdx.x * 32;     // 100000 = 3125*32, exact

    // --- A-tile fill via CDNA5 async copy: global -> LDS, no VGPR staging ---
    // 32 rows x 512 cols (agg cols 0-255, x cols 256-511); 4096 b128 moves.
#pragma unroll
    for (int t = 0; t < 16; ++t) {
        int v  = tid + t * 256;              // float4 index, 0..4095
        int r  = v >> 7;                     // row 0..31
        int cv = v & 127;                    // float4 col within 512
        const float* srcp = (cv < 64)
            ? (agg + (size_t)(rowBase + r) * D + cv * 4)
            : (x   + (size_t)(rowBase + r) * D + (cv - 64) * 4);
        // low 32 bits of a generic shared-memory address == LDS byte offset
        unsigned ldsoff = (unsigned)(unsigned long long)(uintptr_t)&At[r * LDA + cv * 4];
        asm volatile("global_load_async_to_lds_b128 %0, %1, off"
                     :: "v"(ldsoff), "v"(srcp) : "memory");
    }
    asm volatile("s_wait_asynccnt 0" ::: "memory");
    __syncthreads();

    const int wave    = tid >> 5;
    const int lane    = tid & 31;
    const int laneN   = lane & 15;
    const int laneHi  = lane >> 4;           // 0: K=k,k+1   1: K=k+2,k+3
    const int bcol    = wave * 16 + laneN;   // this lane's output column

    const float* aRow0 = &At[laneN * LDA + 2 * laneHi];        // row tile 0 (rows 0-15)
    const float* aRow1 = aRow0 + 16 * LDA;                     // row tile 1 (rows 16-31)
    const float* Wr    = Wrel  + bcol;
    const float* Wt    = Wroot + bcol;

    // --- GEMM1: K = 512 via V_WMMA_F32_16X16X4_F32, B frag reused across 2 M-tiles ---
    v8f acc0 = {}, acc1 = {};
#pragma unroll 4
    for (int k = 0; k < 256; k += 4) {
        int kk = k + 2 * laneHi;
        v2f b; b.x = Wr[kk * O]; b.y = Wr[(kk + 1) * O];
        v2f a0 = *reinterpret_cast<const v2f*>(aRow0 + k);     // ds_load_b64
        v2f a1 = *reinterpret_cast<const v2f*>(aRow1 + k);
        acc0 = __builtin_amdgcn_wmma_f32_16x16x4_f32(
            false, a0, false, b, (short)0, acc0, false, false);
        acc1 = __builtin_amdgcn_wmma_f32_16x16x4_f32(
            false, a1, false, b, (short)0, acc1, false, false);
    }
#pragma unroll 4
    for (int k = 0; k < 256; k += 4) {
        int kk = k + 2 * laneHi;
        v2f b; b.x = Wt[kk * O]; b.y = Wt[(kk + 1) * O];
        v2f a0 = *reinterpret_cast<const v2f*>(aRow0 + 256 + k);
        v2f a1 = *reinterpret_cast<const v2f*>(aRow1 + 256 + k);
        acc0 = __builtin_amdgcn_wmma_f32_16x16x4_f32(
            false, a0, false, b, (short)0, acc0, false, false);
        acc1 = __builtin_amdgcn_wmma_f32_16x16x4_f32(
            false, a1, false, b, (short)0, acc1, false, false);
    }

    // --- bias + relu, stage h tile (32x128) in LDS ---
    // C/D layout: VGPR i -> row i (lanes 0-15) / row i+8 (lanes 16-31), col = lane&15
    const float br = brel[bcol];
#pragma unroll
    for (int i = 0; i < 8; ++i) {
        float h0 = acc0[i] + br;
        float h1 = acc1[i] + br;
        h0 = h0 > 0.f ? h0 : 0.f;
        h1 = h1 > 0.f ? h1 : 0.f;
        Ht[(i + 8 * laneHi) * LDH + bcol]        = h0;
        Ht[(16 + i + 8 * laneHi) * LDH + bcol]   = h1;
    }
    __syncthreads();

    // --- GEMM2: K = 128, same two-row-tile structure ---
    v8f mu0 = {}, mu1 = {};
    const float* hRow0 = &Ht[laneN * LDH + 2 * laneHi];
    const float* hRow1 = hRow0 + 16 * LDH;
    const float* Wm    = Wmu + bcol;
#pragma unroll 4
    for (int k = 0; k < 128; k += 4) {
        int kk = k + 2 * laneHi;
        v2f b; b.x = Wm[kk * O]; b.y = Wm[(kk + 1) * O];
        v2f a0 = *reinterpret_cast<const v2f*>(hRow0 + k);
        v2f a1 = *reinterpret_cast<const v2f*>(hRow1 + k);
        mu0 = __builtin_amdgcn_wmma_f32_16x16x4_f32(
            false, a0, false, b, (short)0, mu0, false, false);
        mu1 = __builtin_amdgcn_wmma_f32_16x16x4_f32(
            false, a1, false, b, (short)0, mu1, false, false);
    }

    // --- bias + relu + eval-mode rrelu (identity on relu output, kept for fidelity) ---
    const float bm = bmu[bcol];
#pragma unroll
    for (int i = 0; i < 8; ++i) {
        float m0 = mu0[i] + bm;
        float m1 = mu1[i] + bm;
        m0 = m0 > 0.f ? m0 : 0.f;
        m1 = m1 > 0.f ? m1 : 0.f;
        float o0 = m0 >= 0.f ? m0 : m0 * RRELU_SLOPE;
        float o1 = m1 >= 0.f ? m1 : m1 * RRELU_SLOPE;
        out[(size_t)(rowBase + i + 8 * laneHi) * O + bcol]      = o0;
        out[(size_t)(rowBase + 16 + i + 8 * laneHi) * O + bcol] = o1;
    }
}

// ---------------- second tuple output: target_edge_weights passthrough ----------------
__global__ void tail_kernel(const float* __restrict__ tw, float* __restrict__ out) {
    int i = blockIdx.x * blockDim.x + threadIdx.x;
    if (i < E_TGT) out[(size_t)N_NODES * O + i] = tw[i];
}

extern "C" void kernel_launch(void* const* d_in, const int* in_sizes, int n_in,
                              void* d_out, int out_size, void* d_ws, size_t ws_size,
                              hipStream_t stream) {
    const float* x     = (const float*)d_in[0];
    const int*   mei   = (const int*)d_in[1];
    const int*   tei   = (const int*)d_in[2];
    const float* tw    = (const float*)d_in[3];
    const float* Wrel  = (const float*)d_in[4];
    const float* brel  = (const float*)d_in[5];
    const float* Wroot = (const float*)d_in[6];
    const float* Wmu   = (const float*)d_in[7];
    const float* bmu   = (const float*)d_in[8];
    float* out = (float*)d_out;
    float* agg = (float*)d_ws;             // 100000 x 256 fp32 = 102.4 MB scratch

    const int n4 = N_NODES * D / 4;        // 6.4M float4
    zero_kernel<<<(n4 + 255) / 256, 256, 0, stream>>>((float4*)agg, n4);

    const int nblk_scatter = (E_MSG + E_TGT + 3) / 4;   // 402,500
    scatter_kernel<<<nblk_scatter, 256, 0, stream>>>(x, mei, tei, tw, agg);

    fused_gemm_kernel<<<N_NODES / 32, 256, 0, stream>>>(
        agg, x, Wrel, brel, Wroot, Wmu, bmu, out);

    tail_kernel<<<(E_TGT + 255) / 256, 256, 0, stream>>>(tw, out);
}